// EMB_71442486001720
// MI455X (gfx1250) — compile-verified
//
#include <hip/hip_runtime.h>

// ---------------------------------------------------------------------------
// GCN layer for MI455X (gfx1250, wave32):
//   support = x @ W              -> fp32 WMMA (V_WMMA_F32_16X16X4_F32)
//   msgs    = support[src] * w   -> coalesced gather (L2-resident, 192MB L2)
//   agg     = segment_sum(dst)   -> global_atomic_add_f32 into d_out
//   out     = relu(agg + b)      -> in-place epilogue
// Streamed-once data (x, edge lists) uses non-temporal loads so the
// support/out working set (~102 MB) stays resident in the 192 MB L2.
// ---------------------------------------------------------------------------

typedef __attribute__((ext_vector_type(2))) float v2f;
typedef __attribute__((ext_vector_type(8))) float v8f;

#define NFEAT 256
#define NHID  128
#define KC    64            // K-chunk of W staged in LDS
#define PR_STRIDE 288       // dwords per K-pair row: 128*2 + 32 pad (bank-conflict free)

// WMMA fp32 VGPR layout (ISA 7.12.2), wave32:
//   A (16x4):  lane L -> M = L%16, K = v + (L/16)*2   (v = VGPR index, v2f)
//   B (4x16):  lane L -> N = L%16, K = v + (L/16)*2
//   C/D(16x16):lane L, VGPR v -> M = v + (L/16)*8, N = L%16  (v8f)
//
// LDS layout for W chunk: K-pair interleaved [KC/2][NHID][2] + pad, so a
// B fragment {K, K+1} is one aligned ds_load_b64 into an even VGPR pair.
__global__ __launch_bounds__(256) void gcn_gemm_wmma(
    const float* __restrict__ x, const float* __restrict__ W,
    float* __restrict__ support, int m_tiles, int n_nodes)
{
    __shared__ float ldsW[(KC / 2) * PR_STRIDE];   // 36 KB per block

    const int lane   = threadIdx.x & 31;
    const int wave   = threadIdx.x >> 5;
    const int mt     = blockIdx.x * 8 + wave;      // one 16-row tile per wave
    const bool active = (mt < m_tiles);
    const int mt_c   = active ? mt : 0;

    const int m_lo  = lane & 15;                   // 0..15
    const int khalf = (lane >> 4) * 2;             // 0 or 2

    v8f acc[8] = {};                               // 16x128 fp32 accumulators

    int arow = mt_c * 16 + m_lo;
    if (arow >= n_nodes) arow = n_nodes - 1;
    const float* xrow = x + (long long)arow * NFEAT;

    #pragma unroll 1
    for (int kc = 0; kc < NFEAT; kc += KC) {
        // Stage 64x128 W chunk, K-pair interleaved: (k,n) -> (k/2)*288 + n*2 + (k&1)
        for (int idx = threadIdx.x; idx < KC * NHID; idx += 256) {
            const int k = idx >> 7;                // 0..63
            const int n = idx & (NHID - 1);        // 0..127
            ldsW[(k >> 1) * PR_STRIDE + n * 2 + (k & 1)] = W[kc * NHID + idx];
        }
        __syncthreads();

        if (active) {
            // Preload all 16 A fragments of this chunk (one load clause,
            // single wait), non-temporal: x is streamed exactly once.
            v2f afrag[16];
            #pragma unroll
            for (int i = 0; i < 16; ++i)
                afrag[i] = __builtin_nontemporal_load(
                    (const v2f*)(xrow + kc + i * 4 + khalf));

            #pragma unroll
            for (int kk4 = 0; kk4 < 16; ++kk4) {
                // K-pair row for this lane half: (kk4*4 + khalf)/2
                const float* bbase =
                    &ldsW[(kk4 * 2 + (khalf >> 1)) * PR_STRIDE + m_lo * 2];
                #pragma unroll
                for (int n = 0; n < 8; ++n) {
                    v2f b = *(const v2f*)(bbase + n * 32);   // ds_load_b64
                    acc[n] = __builtin_amdgcn_wmma_f32_16x16x4_f32(
                        /*neg_a=*/false, afrag[kk4], /*neg_b=*/false, b,
                        /*c_mod=*/(short)0, acc[n],
                        /*reuse_a=*/false, /*reuse_b=*/false);
                }
            }
        }
        __syncthreads();
    }

    if (active) {
        const int r_hi = (lane >> 4) * 8;
        #pragma unroll
        for (int n = 0; n < 8; ++n) {
            #pragma unroll
            for (int v = 0; v < 8; ++v) {
                const long long r = (long long)mt * 16 + v + r_hi;
                if (r < n_nodes)
                    support[r * NHID + n * 16 + m_lo] = acc[n][v];
            }
        }
    }
}

// One edge per 128 threads: coalesced 512B gather of the source row,
// scaled, scattered with hardware f32 atomics (support + out live in L2).
// Edge lists are streamed once -> non-temporal loads.
__global__ __launch_bounds__(256) void gcn_edge_scatter(
    const float* __restrict__ support,
    const long long* __restrict__ esrc,
    const long long* __restrict__ edst,
    const float* __restrict__ ew,
    float* __restrict__ out, int n_edges)
{
    const int e = blockIdx.x * 2 + (threadIdx.x >> 7);
    const int c = threadIdx.x & (NHID - 1);
    if (e < n_edges) {
        const long long s = __builtin_nontemporal_load(esrc + e);
        const long long d = __builtin_nontemporal_load(edst + e);
        const float     w = __builtin_nontemporal_load(ew + e);
        const float v = support[s * NHID + c] * w;
        __hip_atomic_fetch_add(out + d * NHID + c, v,
                               __ATOMIC_RELAXED, __HIP_MEMORY_SCOPE_AGENT);
    }
}

__global__ __launch_bounds__(256) void gcn_bias_relu(
    float* __restrict__ out, const float* __restrict__ bias, long long total)
{
    const long long i = (long long)blockIdx.x * 256 + threadIdx.x;
    if (i < total) {
        const float v = out[i] + bias[i & (NHID - 1)];
        out[i] = v > 0.0f ? v : 0.0f;
    }
}

extern "C" void kernel_launch(void* const* d_in, const int* in_sizes, int n_in,
                              void* d_out, int out_size, void* d_ws, size_t ws_size,
                              hipStream_t stream) {
    const float*      x    = (const float*)d_in[0];
    const long long*  esrc = (const long long*)d_in[1];   // int64 edge indices
    const long long*  edst = (const long long*)d_in[2];
    const float*      ew   = (const float*)d_in[3];
    const float*      W    = (const float*)d_in[4];
    const float*      bias = (const float*)d_in[5];
    float*            out  = (float*)d_out;
    float*            support = (float*)d_ws;             // n_nodes*128 f32 (51.2 MB)

    const int n_nodes = in_sizes[0] / NFEAT;
    const int n_edges = in_sizes[1];
    const int m_tiles = (n_nodes + 15) / 16;
    const long long total = (long long)n_nodes * NHID;

    // Zero the aggregation buffer every call (graph-capturable memset node).
    hipMemsetAsync(out, 0, (size_t)total * sizeof(float), stream);

    gcn_gemm_wmma<<<(m_tiles + 7) / 8, 256, 0, stream>>>(x, W, support,
                                                         m_tiles, n_nodes);

    gcn_edge_scatter<<<(n_edges + 1) / 2, 256, 0, stream>>>(support, esrc, edst,
                                                            ew, out, n_edges);

    gcn_bias_relu<<<(unsigned)((total + 255) / 256), 256, 0, stream>>>(out, bias,
                                                                       total);
}